// Local_Diffusion_36567351558725
// MI455X (gfx1250) — compile-verified
//
#include <hip/hip_runtime.h>

typedef float v2f __attribute__((ext_vector_type(2)));
typedef float v8f __attribute__((ext_vector_type(8)));

#define WMMA_F32_16X16X4(A, B, C) \
  __builtin_amdgcn_wmma_f32_16x16x4_f32(false, (A), false, (B), (short)0, (C), false, false)

// ---------------------------------------------------------------------------
// 1) deg[n] = 1.0 (self-loop weight)
// ---------------------------------------------------------------------------
__global__ void gcn_deg_init(float* __restrict__ deg, int N) {
  int i = blockIdx.x * blockDim.x + threadIdx.x;
  if (i < N) deg[i] = 1.0f;
}

// ---------------------------------------------------------------------------
// 2) deg[col[e]] += edge_attr[e]   (hardware f32 atomics)
// ---------------------------------------------------------------------------
__global__ void gcn_deg_accum(const long long* __restrict__ ei,
                              const float* __restrict__ ea,
                              float* __restrict__ deg, int E) {
  int e = blockIdx.x * blockDim.x + threadIdx.x;
  if (e >= E) return;
  int c = (int)ei[(long long)E + e];  // target node (node ids < 2^31)
  unsafeAtomicAdd(&deg[c], ea[e]);
}

// ---------------------------------------------------------------------------
// 3) dis[n] = deg>0 ? rsqrt(deg) : 0   (in place)
// ---------------------------------------------------------------------------
__global__ void gcn_deg_to_dis(float* __restrict__ deg, int N) {
  int i = blockIdx.x * blockDim.x + threadIdx.x;
  if (i >= N) return;
  float d = deg[i];
  deg[i] = (d > 0.0f) ? rsqrtf(d) : 0.0f;
}

// ---------------------------------------------------------------------------
// 4) xt = x @ W  via V_WMMA_F32_16X16X4_F32 (two K=4 chunks for K=8).
//    One wave computes a 16-row x 64-col slab: 4 tiles x 2 WMMA = 8 wmma ops.
//    A 16x4 layout: lanes 0-15 row M, VGPR{0,1}=K{0,1}; lanes 16-31 K{2,3}.
//    B 4x16 layout: VGPR0 = rows K0 (lanes0-15)/K2 (lanes16-31), VGPR1 = K1/K3.
//    C/D 16x16: VGPR v = row base+v (lanes 0-15) / base+8+v (lanes 16-31).
//    Tail handling is a wave-uniform scalar branch so EXEC stays all-ones
//    around the WMMAs and the hot path is 8 straight stores w/ imm offsets.
// ---------------------------------------------------------------------------
__global__ void __launch_bounds__(256)
gcn_gemm_wmma(const float* __restrict__ x, const float* __restrict__ W,
              float* __restrict__ xt, int N, int nTiles) {
  const int lane = threadIdx.x & 31;
  const int wave = (int)((blockIdx.x * blockDim.x + threadIdx.x) >> 5);
  if (wave >= nTiles) return;  // wave-uniform: EXEC all-ones for WMMA

  const int m  = lane & 15;    // row-in-tile (A) / col-in-tile (B,D)
  const int hi = lane >> 4;    // 0 or 1
  const int kh = hi * 2;       // K offset within a 4-chunk

  const int rowBase = wave * 16;
  int row = rowBase + m;
  if (row >= N) row = N - 1;   // clamp loads (no tail for N=100000)

  const float* xr = x + (long long)row * 8;
  v2f aA, aB;
  aA[0] = xr[kh + 0];     aA[1] = xr[kh + 1];      // K = 0..3
  aB[0] = xr[4 + kh + 0]; aB[1] = xr[4 + kh + 1];  // K = 4..7

  const int mBase = rowBase + hi * 8;
  float* pBase = xt + (long long)mBase * 64;       // one 64-bit addr, reused
  const bool fullTile = (rowBase + 16 <= N);       // wave-uniform

#pragma unroll
  for (int j = 0; j < 4; ++j) {
    const int col = j * 16 + m;
    v2f bA, bB;
    bA[0] = W[(kh + 0) * 64 + col];     bA[1] = W[(kh + 1) * 64 + col];
    bB[0] = W[(4 + kh + 0) * 64 + col]; bB[1] = W[(4 + kh + 1) * 64 + col];

    v8f c = {};
    c = WMMA_F32_16X16X4(aA, bA, c);
    c = WMMA_F32_16X16X4(aB, bB, c);

    float* p = pBase + col;
    if (fullTile) {
#pragma unroll
      for (int v = 0; v < 8; ++v) p[v * 64] = c[v];   // imm-offset stores
    } else {
#pragma unroll
      for (int v = 0; v < 8; ++v)
        if (mBase + v < N) p[v * 64] = c[v];
    }
  }
}

// ---------------------------------------------------------------------------
// 5) out[n,f] = b[f] + xt[n,f] * dis[n]^2   (bias + self-loop message)
// ---------------------------------------------------------------------------
__global__ void gcn_init_out(const float* __restrict__ xt,
                             const float* __restrict__ dis,
                             const float* __restrict__ b,
                             float* __restrict__ out, int n64) {
  int t = blockIdx.x * blockDim.x + threadIdx.x;
  if (t >= n64) return;
  int n = t >> 6;
  int f = t & 63;
  float d = dis[n];
  out[t] = b[f] + xt[t] * d * d;
}

// ---------------------------------------------------------------------------
// 6) out[col] += xt[row] * (dis[row]*ew*dis[col]); 64 threads per edge.
//    Lanes of a wave span contiguous features of one edge, so the scalar
//    loads are wave-uniform (L1 broadcast) and xt/out accesses are
//    128B-coalesced. unsafeAtomicAdd -> global_atomic_add_f32 (L2-resident:
//    out + xt = 51 MB fits the 192 MB L2).
// ---------------------------------------------------------------------------
__global__ void gcn_scatter(const long long* __restrict__ ei,
                            const float* __restrict__ ea,
                            const float* __restrict__ dis,
                            const float* __restrict__ xt,
                            float* __restrict__ out, int E) {
  long long t = (long long)blockIdx.x * blockDim.x + threadIdx.x;
  int e = (int)(t >> 6);
  if (e >= E) return;
  int f = (int)(t & 63);
  int r = (int)ei[e];
  int c = (int)ei[(long long)E + e];
  float nrm = dis[r] * ea[e] * dis[c];
  // 32-bit offsets: max index = 100000*64 < 2^31
  unsafeAtomicAdd(&out[c * 64 + f], xt[r * 64 + f] * nrm);
}

// ---------------------------------------------------------------------------
extern "C" void kernel_launch(void* const* d_in, const int* in_sizes, int n_in,
                              void* d_out, int out_size, void* d_ws, size_t ws_size,
                              hipStream_t stream) {
  const float* x      = (const float*)d_in[0];      // [N, 8]
  const long long* ei = (const long long*)d_in[1];  // [2, E] int64
  const float* ea     = (const float*)d_in[2];      // [E]
  const float* W      = (const float*)d_in[3];      // [8, 64]
  const float* b      = (const float*)d_in[4];      // [64]
  float* out          = (float*)d_out;              // [N, 64]

  const int N = in_sizes[0] / 8;
  const int E = in_sizes[2];

  // Workspace layout: dis [N floats] | xt [N*64 floats]  (~26 MB)
  float* deg = (float*)d_ws;
  size_t degBytes = (((size_t)N * sizeof(float)) + 255) & ~(size_t)255;
  float* xt = (float*)((char*)d_ws + degBytes);

  const int T = 256;

  gcn_deg_init<<<(N + T - 1) / T, T, 0, stream>>>(deg, N);
  gcn_deg_accum<<<(E + T - 1) / T, T, 0, stream>>>(ei, ea, deg, E);
  gcn_deg_to_dis<<<(N + T - 1) / T, T, 0, stream>>>(deg, N);

  const int nTiles = (N + 15) / 16;                 // 6250 waves
  const int wavesPerBlock = T / 32;                 // 8
  const int gemmBlocks = (nTiles + wavesPerBlock - 1) / wavesPerBlock;
  gcn_gemm_wmma<<<gemmBlocks, T, 0, stream>>>(x, W, xt, N, nTiles);

  const long long n64 = (long long)N * 64;
  gcn_init_out<<<(unsigned)((n64 + T - 1) / T), T, 0, stream>>>(xt, deg, b, out, (int)n64);

  const long long e64 = (long long)E * 64;
  gcn_scatter<<<(unsigned)((e64 + T - 1) / T), T, 0, stream>>>(ei, ea, deg, xt, out, E);
}